// FeedForwardQuantum_65481071403399
// MI455X (gfx1250) — compile-verified
//
#include <hip/hip_runtime.h>
#include <hip/hip_bf16.h>
#include <math.h>

// ---------------------------------------------------------------------------
// Problem constants: B=8, T=2048, E=1024, F=4096, NQ=10 -> N = 16384 tokens
// ---------------------------------------------------------------------------
#define NTOK   16384
#define EDIM   1024
#define FDIM   4096
#define NQ     10
#define KQ     12            // NQ padded to a multiple of 4 (zero-filled)

typedef float v2f __attribute__((ext_vector_type(2)));
typedef float v8f __attribute__((ext_vector_type(8)));
typedef int   v4i __attribute__((vector_size(16)));   // matches builtin param type

// --- CDNA5 async global->LDS path (probe via __has_builtin) ----------------
#if __has_builtin(__builtin_amdgcn_global_load_async_to_lds_b128)
#define HAVE_ASYNC 1
typedef __attribute__((address_space(1))) v4i* gvec_t;   // global src (non-const)
typedef __attribute__((address_space(3))) v4i* lvec_t;   // LDS dst
#else
#define HAVE_ASYNC 0
#endif

#if HAVE_ASYNC
#if __has_builtin(__builtin_amdgcn_s_wait_asynccnt)
#define WAIT_ASYNC(n) __builtin_amdgcn_s_wait_asynccnt(n)
#else
#define WAIT_ASYNC(n) asm volatile("s_wait_asynccnt %0" ::"i"(n))
#endif
#else
#define WAIT_ASYNC(n) do {} while (0)
#endif

// ---------------------------------------------------------------------------
// Kernel 1: quantum layer. One wave32 per token; 32 amplitudes per lane in
// registers (amp[r] = state[r*32 + lane]).  Wire q <-> bit (9-q).
// params-RY layer + CNOT ring are token independent; the CNOT ring is a pure
// basis permutation P, so psi0[j] = prod_q (cos|sin)(params_q/2) at bits of
// P^{-1}(j).  Per token: 10 RY(x_q) butterflies + <Z_q> reductions.
// ---------------------------------------------------------------------------
__global__ __launch_bounds__(256)
void qsim_kernel(const float* __restrict__ x,      // [NTOK, EDIM] (first NQ cols used)
                 const float* __restrict__ params, // [2, NQ] (row 0 used)
                 float* __restrict__ qout)         // [NTOK, NQ]
{
    const int lane  = threadIdx.x & 31;
    const int wid   = threadIdx.x >> 5;
    const int token = blockIdx.x * 8 + wid;

    float cp[NQ], sp[NQ];
#pragma unroll
    for (int q = 0; q < NQ; ++q) {
        const float h = 0.5f * params[q];
        cp[q] = __cosf(h);
        sp[q] = __sinf(h);
    }
    float xc[NQ], xs[NQ];
#pragma unroll
    for (int q = 0; q < NQ; ++q) {
        const float h = 0.5f * x[(size_t)token * EDIM + q];
        xc[q] = __cosf(h);
        xs[q] = __sinf(h);
    }

    // psi0 via analytic product + inverse CNOT permutation
    float amp[32];
#pragma unroll
    for (int r = 0; r < 32; ++r) {
        int b = (r << 5) | lane;
        b ^= (b & 1) << 9;                          // undo C(9,0)
#pragma unroll
        for (int q = 8; q >= 0; --q)                // undo C(q,q+1)
            b ^= ((b >> (9 - q)) & 1) << (8 - q);
        float v = 1.0f;
#pragma unroll
        for (int q = 0; q < NQ; ++q)
            v *= ((b >> (9 - q)) & 1) ? sp[q] : cp[q];
        amp[r] = v;
    }

    // wires 0..4: register-bit butterflies
#pragma unroll
    for (int q = 0; q < 5; ++q) {
        const int m = 1 << (4 - q);
        const float c = xc[q], s = xs[q];
#pragma unroll
        for (int r = 0; r < 32; ++r) {
            if (!(r & m)) {
                const float a0 = amp[r], a1 = amp[r + m];
                amp[r]     = c * a0 - s * a1;
                amp[r + m] = s * a0 + c * a1;
            }
        }
    }
    // wires 5..9: lane-bit butterflies
#pragma unroll
    for (int q = 5; q < NQ; ++q) {
        const int m = 1 << (9 - q);
        const float c = xc[q], s = xs[q];
        const bool hi = (lane & m) != 0;
#pragma unroll
        for (int r = 0; r < 32; ++r) {
            const float p = __shfl_xor(amp[r], m, 32);
            amp[r] = hi ? (s * p + c * amp[r]) : (c * amp[r] - s * p);
        }
    }

    // probabilities and <Z_q>
    float ptot = 0.0f;
#pragma unroll
    for (int r = 0; r < 32; ++r) {
        amp[r] = amp[r] * amp[r];
        ptot += amp[r];
    }
    float z[NQ];
#pragma unroll
    for (int q = 0; q < 5; ++q) {
        const int m = 1 << (4 - q);
        float a = 0.0f;
#pragma unroll
        for (int r = 0; r < 32; ++r)
            a += (r & m) ? -amp[r] : amp[r];
        z[q] = a;
    }
#pragma unroll
    for (int q = 5; q < NQ; ++q)
        z[q] = (lane & (1 << (9 - q))) ? -ptot : ptot;

#pragma unroll
    for (int q = 0; q < NQ; ++q) {
#pragma unroll
        for (int off = 16; off > 0; off >>= 1)
            z[q] += __shfl_xor(z[q], off, 32);
    }
    if (lane == 0) {
#pragma unroll
        for (int q = 0; q < NQ; ++q)
            qout[(size_t)token * NQ + q] = z[q];
    }
}

// ---------------------------------------------------------------------------
// Kernel 2: fused  out = relu(q @ w1^T) @ w2^T  entirely on the WMMA pipe.
// Block tile 64(M) x 128(N), K-chunk 32.
//  * h tile (64x32) is itself computed with WMMA: each of 8 waves does one
//    16x16 tile = 3 x v_wmma (K = 10 padded to 12, zeros), relu on the C
//    registers, stored to LDS in the A-fragment layout.  q A-fragments are
//    loop-invariant (hoisted).
//  * main GEMM: 8 waves as 2(M) x 4(N), each 32x32 strip = 2x2 WMMA tiles,
//    8 K-steps/chunk; A/B pairs are single b64 LDS loads.
//  * w2 chunks double-buffered in LDS ([n][k] rows padded to 36 floats =
//    144B, 16B-aligned), filled by async global->LDS b128 DMA (ASYNCcnt).
// ---------------------------------------------------------------------------
#define TM 64
#define TN 128
#define TK 32
#define NCHUNK (FDIM / TK)   // 128

__global__ __launch_bounds__(256)
void ffn_wmma_kernel(const float* __restrict__ qin, // [NTOK, NQ]
                     const float* __restrict__ w1,  // [FDIM, NQ]
                     const float* __restrict__ w2,  // [EDIM, FDIM]
                     float* __restrict__ out)       // [NTOK, EDIM]
{
    __shared__ float qs [TM][KQ];            // q rows, K padded 10->12 (zeros)
    __shared__ float w1c[2][TK][KQ];         // w1 slice [kk][j], double buffered
    __shared__ float hs [TM][34];            // relu(q.w1^T) tile (rows 8B-aligned)
    __shared__ float w2l[2][TN][36];         // w2 chunk [n][k], rows 144B (16B-aligned)

    const int t    = threadIdx.x;
    const int lane = t & 31;
    const int wid  = t >> 5;
    const int wm   = wid & 1;                // main GEMM: wave M row (32 rows)
    const int wn   = wid >> 1;               // main GEMM: wave N col (32 cols)
    const int hm0  = (wid & 3) * 16;         // h GEMM: wave's 16-row strip
    const int hk0  = (wid >> 2) * 16;        // h GEMM: wave's 16-col strip
    const int mb   = blockIdx.x * TM;
    const int nb   = blockIdx.y * TN;

    // fragment addressing (16x4 f32 A/B layout):
    //   lanes 0-15: K={0,1}, lanes 16-31: K={2,3}; M/N = lane % 16
    const int mm = lane & 15;
    const int kp = (lane >> 4) << 1;

    // ---- w2 chunk fill: 1024 x b128 async DMA, 4 per thread ---------------
    auto issue_w2 = [&](int kb, int buf) {
#pragma unroll
        for (int i = 0; i < 4; ++i) {
            const int v   = t + 256 * i;     // 0..1023 float4 slots
            const int row = v >> 3;          // e offset (0..127)
            const int f4  = v & 7;           // float4 within 32-f chunk
            const float* gp = &w2[(size_t)(nb + row) * FDIM + kb + f4 * 4];
            float*       lp = &w2l[buf][row][f4 * 4];
#if HAVE_ASYNC
            __builtin_amdgcn_global_load_async_to_lds_b128(
                (gvec_t)gp, (lvec_t)lp, /*offset=*/0, /*cpol=*/0);
#else
            *reinterpret_cast<float4*>(lp) = *reinterpret_cast<const float4*>(gp);
#endif
        }
    };
    auto load_w1 = [&](int kb, int buf) {
        for (int i = t; i < TK * KQ; i += 256) {
            const int kk = i / KQ, j = i % KQ;
            w1c[buf][kk][j] = (j < NQ) ? w1[(size_t)(kb + kk) * NQ + j] : 0.0f;
        }
    };

    // ---- prologue ---------------------------------------------------------
    for (int i = t; i < TM * KQ; i += 256) {
        const int m = i / KQ, j = i % KQ;
        qs[m][j] = (j < NQ) ? qin[(size_t)(mb + m) * NQ + j] : 0.0f;
    }
    load_w1(0, 0);
    issue_w2(0, 0);

    v8f acc[2][2];
#pragma unroll
    for (int tm = 0; tm < 2; ++tm)
#pragma unroll
        for (int tn = 0; tn < 2; ++tn)
            acc[tm][tn] = (v8f){0.f, 0.f, 0.f, 0.f, 0.f, 0.f, 0.f, 0.f};

    WAIT_ASYNC(0);
    __syncthreads();

    // q A-fragments for the h GEMM are loop-invariant: hoist (3 x v2f)
    v2f qa[3];
#pragma unroll
    for (int s = 0; s < 3; ++s)
        qa[s] = *reinterpret_cast<const v2f*>(&qs[hm0 + mm][4 * s + kp]);

    // ---- main K loop: double-buffered async prefetch ----------------------
    for (int ic = 0; ic < NCHUNK; ++ic) {
        const int buf = ic & 1;

        if (ic + 1 < NCHUNK) {               // prefetch next chunk
            issue_w2((ic + 1) * TK, buf ^ 1);
            load_w1((ic + 1) * TK, buf ^ 1);
        }

        // h tile on the WMMA pipe: one 16x16 tile per wave, K=12 (3 steps)
        {
            v8f h = (v8f){0.f, 0.f, 0.f, 0.f, 0.f, 0.f, 0.f, 0.f};
#pragma unroll
            for (int s = 0; s < 3; ++s) {
                const v2f b = *reinterpret_cast<const v2f*>(
                    &w1c[buf][hk0 + mm][4 * s + kp]);   // B(k=j, n=kk) pairs
                h = __builtin_amdgcn_wmma_f32_16x16x4_f32(
                    false, qa[s], false, b, (short)0, h, false, false);
            }
            // relu on C registers, store to hs in A-fragment layout
#pragma unroll
            for (int v = 0; v < 8; ++v)
                hs[hm0 + v + 8 * (lane >> 4)][hk0 + mm] = fmaxf(h[v], 0.0f);
        }

        // drain current chunk's DMA (next chunk may stay in flight)
        if (ic + 1 < NCHUNK) { WAIT_ASYNC(4); } else { WAIT_ASYNC(0); }
        __syncthreads();

        // main WMMA: 8 K-steps x (2x2) tiles; A/B pairs are single b64 loads
#pragma unroll
        for (int ks = 0; ks < TK; ks += 4) {
            v2f a[2], b[2];
#pragma unroll
            for (int tm = 0; tm < 2; ++tm)
                a[tm] = *reinterpret_cast<const v2f*>(
                    &hs[wm * 32 + tm * 16 + mm][ks + kp]);
#pragma unroll
            for (int tn = 0; tn < 2; ++tn)
                b[tn] = *reinterpret_cast<const v2f*>(
                    &w2l[buf][wn * 32 + tn * 16 + mm][ks + kp]);
#pragma unroll
            for (int tm = 0; tm < 2; ++tm)
#pragma unroll
                for (int tn = 0; tn < 2; ++tn)
                    acc[tm][tn] = __builtin_amdgcn_wmma_f32_16x16x4_f32(
                        false, a[tm], false, b[tn],
                        (short)0, acc[tm][tn], false, false);
        }
        __syncthreads();
    }

    // ---- store: C/D layout  M = v + 8*(lane/16), N = lane%16 --------------
#pragma unroll
    for (int tm = 0; tm < 2; ++tm) {
        const int r0 = mb + wm * 32 + tm * 16 + (lane >> 4) * 8;
#pragma unroll
        for (int tn = 0; tn < 2; ++tn) {
            const int col = nb + wn * 32 + tn * 16 + mm;
#pragma unroll
            for (int v = 0; v < 8; ++v)
                out[(size_t)(r0 + v) * EDIM + col] = acc[tm][tn][v];
        }
    }
}

// ---------------------------------------------------------------------------
// Launch: inputs in setup_inputs() order: x, params, w1, w2
// ---------------------------------------------------------------------------
extern "C" void kernel_launch(void* const* d_in, const int* in_sizes, int n_in,
                              void* d_out, int out_size, void* d_ws, size_t ws_size,
                              hipStream_t stream)
{
    (void)in_sizes; (void)n_in; (void)out_size; (void)ws_size;

    const float* x      = (const float*)d_in[0];   // [8,2048,1024]
    const float* params = (const float*)d_in[1];   // [2,10]
    const float* w1     = (const float*)d_in[2];   // [4096,10]
    const float* w2     = (const float*)d_in[3];   // [1024,4096]
    float*       out    = (float*)d_out;           // [8,2048,1024]
    float*       qbuf   = (float*)d_ws;            // [16384,10] scratch

    qsim_kernel<<<NTOK / 8, 256, 0, stream>>>(x, params, qbuf);

    dim3 grid(NTOK / TM, EDIM / TN);   // (256, 8)
    ffn_wmma_kernel<<<grid, 256, 0, stream>>>(qbuf, w1, w2, out);
}